// GATLayer_5446018531915
// MI455X (gfx1250) — compile-verified
//
#include <hip/hip_runtime.h>
#include <hip/hip_bf16.h>
#include <math.h>

typedef __attribute__((ext_vector_type(2))) float v2f;
typedef __attribute__((ext_vector_type(8))) float v8f;

// ---------------------------------------------------------------------------
// Generic fp32 WMMA GEMM: C[M,N] = A[M,K] @ B[K,N]
//   - K % 4 == 0, N == NT*16, K*N <= 8192 (B staged pair-packed in LDS)
//   - one wave computes a 16 x N output tile via V_WMMA_F32_16X16X4_F32
// A-frag (16x4): lane L holds row M=L&15, K pair at k0 + 2*(L>>4)
// B-frag (4x16): lane L holds col N=L&15, same K pair
// C/D (16x16):   reg r, lane L -> row r + 8*(L>>4), col L&15
// LDS holds B as K-pairs: sB2[(k>>1)*N + n] = {B[k][n], B[k+1][n]} (k even),
// so each lane's B fragment is a single 8B ds_load_b64 into an even VGPR pair.
// ---------------------------------------------------------------------------
template <int NT>
__global__ void __launch_bounds__(256) gemm_f32_wmma(
    const float* __restrict__ A, const float* __restrict__ B,
    float* __restrict__ C, int M, int K, int N)
{
    __shared__ v2f sB2[4096];                  // 32 KB
    const int tid = threadIdx.x;
    const int pairCount = (K >> 1) * N;
    for (int p = tid; p < pairCount; p += 256) {
        const int kk = (p / N) << 1;
        const int n  = p - (p / N) * N;
        v2f b;
        b.x = B[kk * N + n];
        b.y = B[(kk + 1) * N + n];
        sB2[p] = b;
    }
    __syncthreads();

    const int lane    = tid & 31;
    const int wave    = tid >> 5;
    const int rowTile = blockIdx.x * 8 + wave;
    const int numTiles = (M + 15) >> 4;
    if (rowTile >= numTiles) return;          // wave-uniform exit (EXEC stays full)

    const int half = lane >> 4;               // 0 or 1
    const int l15  = lane & 15;
    int mrow = rowTile * 16 + l15;
    if (mrow >= M) mrow = M - 1;              // duplicate-read guard (stores masked below)

    v8f acc[NT] = {};
    const float* arow = A + (size_t)mrow * K + 2 * half;
    const v2f* __restrict__ bcol = sB2 + half * N + l15;   // pair row (k0>>1)+half, col l15

    for (int k0 = 0; k0 < K; k0 += 4) {
        const v2f a = *(const v2f*)(arow + k0);
        const v2f* brow = bcol + (k0 >> 1) * N;
#pragma unroll
        for (int nt = 0; nt < NT; ++nt) {
            const v2f b = brow[nt * 16];      // single ds_load_b64
            acc[nt] = __builtin_amdgcn_wmma_f32_16x16x4_f32(
                false, a, false, b, (short)0, acc[nt], false, false);
        }
    }

#pragma unroll
    for (int nt = 0; nt < NT; ++nt) {
#pragma unroll
        for (int r = 0; r < 8; ++r) {
            const int row = rowTile * 16 + r + 8 * half;
            if (row < M)
                C[(size_t)row * N + nt * 16 + l15] = acc[nt][r];
        }
    }
}

// ---------------------------------------------------------------------------
__device__ __forceinline__ void atomicMaxFloat(float* addr, float val) {
    const int iv = __float_as_int(val);
    if (iv >= 0) atomicMax((int*)addr, iv);               // GLOBAL_ATOMIC_MAX_I32
    else         atomicMin((unsigned int*)addr, (unsigned int)iv); // MIN_U32
}

// per-node attention scores: s_src = z.w_attn[0:64], s_dst = z.w_attn[96:160]
__global__ void node_score_kernel(const float* __restrict__ z,
                                  const float* __restrict__ w_attn,
                                  float* __restrict__ ssrc, float* __restrict__ sdst,
                                  int N)
{
    const int n = blockIdx.x * blockDim.x + threadIdx.x;
    if (n >= N) return;
    const float* zr = z + (size_t)n * 64;
    float s0 = 0.f, s1 = 0.f;
#pragma unroll
    for (int i = 0; i < 64; ++i) {
        const float v = zr[i];
        s0 += v * w_attn[i];
        s1 += v * w_attn[96 + i];
    }
    ssrc[n] = s0;
    sdst[n] = s1;
}

__global__ void init_kernel(float* __restrict__ h, float* __restrict__ mbuf,
                            float* __restrict__ den, float* __restrict__ zphi,
                            int hsize, int N)
{
    const int i = blockIdx.x * blockDim.x + threadIdx.x;
    const int stride = gridDim.x * blockDim.x;
    for (int j = i; j < hsize; j += stride) h[j] = 0.f;
    for (int j = i; j < N; j += stride) { mbuf[j] = -INFINITY; den[j] = 0.f; }
    if (i < 256) zphi[i] = 0.f;
}

// a_e = leaky_relu(s_src[src] + eft.w_mid + s_dst[dst]); segment-max into m[dst]
__global__ void edge_score_kernel(const float* __restrict__ eft,
                                  const int* __restrict__ src, const int* __restrict__ dst,
                                  const float* __restrict__ ssrc, const float* __restrict__ sdst,
                                  const float* __restrict__ w_attn,
                                  float* __restrict__ a, float* __restrict__ mbuf, int E)
{
    const int e = blockIdx.x * blockDim.x + threadIdx.x;
    if (e >= E) return;
    const float* er = eft + (size_t)e * 32;
    float t = ssrc[src[e]] + sdst[dst[e]];
#pragma unroll
    for (int i = 0; i < 32; ++i) t += er[i] * w_attn[64 + i];
    const float lr = t > 0.f ? t : 0.01f * t;   // jax leaky_relu slope = 0.01
    a[e] = lr;
    atomicMaxFloat(&mbuf[dst[e]], lr);
}

// ex = exp(a - m[dst]); segment-sum into den[dst]
__global__ void edge_exp_kernel(const int* __restrict__ dst,
                                const float* __restrict__ mbuf,
                                float* __restrict__ a, float* __restrict__ den, int E)
{
    const int e = blockIdx.x * blockDim.x + threadIdx.x;
    if (e >= E) return;
    const int d = dst[e];
    const float ex = __expf(a[e] - mbuf[d]);
    a[e] = ex;
    atomicAdd(&den[d], ex);
}

// h[dst*T+etype][0:96] += alpha * [z[src](64) | eft(32)]   (one wave per edge)
__global__ void edge_scatter_kernel(const float* __restrict__ z, const float* __restrict__ eft,
                                    const float* __restrict__ a, const float* __restrict__ den,
                                    const int* __restrict__ src, const int* __restrict__ dst,
                                    const int* __restrict__ etype,
                                    float* __restrict__ h, int E)
{
    const int e    = (int)((blockIdx.x * blockDim.x + threadIdx.x) >> 5);
    const int lane = threadIdx.x & 31;
    if (e >= E) return;
    const int d = dst[e];
    const float alpha = a[e] / den[d];
    const size_t base = ((size_t)d * 4 + etype[e]) * 96;
    const float* zr = z + (size_t)src[e] * 64;
    const float* er = eft + (size_t)e * 32;
#pragma unroll
    for (int j = lane; j < 96; j += 32) {
        const float v = (j < 64) ? zr[j] : er[j - 64];
        atomicAdd(&h[base + j], alpha * v);     // GLOBAL_ATOMIC_ADD_F32
    }
}

// zphi[t*64+o] = sum_n h2[n*256 + t*64+o]   (h2 viewed as [N][T*64])
__global__ void zphi_kernel(const float* __restrict__ h2, float* __restrict__ zphi, int nodes)
{
    const int s = threadIdx.x;                  // 0..255 -> (t,o)
    float sum = 0.f;
    for (int n = blockIdx.x; n < nodes; n += gridDim.x)
        sum += h2[(size_t)n * 256 + s];
    atomicAdd(&zphi[s], sum);
}

__global__ void beta_kernel(const float* __restrict__ zphi,
                            const float* __restrict__ w_sem, float* __restrict__ beta)
{
    if (threadIdx.x != 0) return;
    float wp[4], mx = -INFINITY;
#pragma unroll
    for (int t = 0; t < 4; ++t) {
        float d = 0.f;
        for (int o = 0; o < 64; ++o) d += zphi[t * 64 + o] * w_sem[o];
        const float lr = d > 0.f ? d : 0.01f * d;
        wp[t] = lr;
        mx = fmaxf(mx, lr);
    }
    float s = 0.f;
#pragma unroll
    for (int t = 0; t < 4; ++t) { wp[t] = __expf(wp[t] - mx); s += wp[t]; }
#pragma unroll
    for (int t = 0; t < 4; ++t) beta[t] = wp[t] / s;
}

// Z[n][o] = sum_t beta[t] * h2[n*256 + t*64 + o]
__global__ void out_kernel(const float* __restrict__ h2, const float* __restrict__ beta,
                           float* __restrict__ Z, int N)
{
    const int i = blockIdx.x * blockDim.x + threadIdx.x;
    if (i >= N * 64) return;
    const int n = i >> 6, o = i & 63;
    const float* r = h2 + (size_t)n * 256 + o;
    Z[i] = beta[0] * r[0] + beta[1] * r[64] + beta[2] * r[128] + beta[3] * r[192];
}

// ---------------------------------------------------------------------------
extern "C" void kernel_launch(void* const* d_in, const int* in_sizes, int n_in,
                              void* d_out, int out_size, void* d_ws, size_t ws_size,
                              hipStream_t stream)
{
    const float* node   = (const float*)d_in[0];
    const float* edgef  = (const float*)d_in[1];
    const int*   src    = (const int*)d_in[2];
    const int*   dst    = (const int*)d_in[3];
    const int*   etype  = (const int*)d_in[4];
    const float* w_n    = (const float*)d_in[5];
    const float* w_e    = (const float*)d_in[6];
    const float* w_attn = (const float*)d_in[7];
    const float* w_fc2  = (const float*)d_in[8];
    const float* w_sem  = (const float*)d_in[9];
    float* out = (float*)d_out;

    const int N = in_sizes[0] / 128;   // 50000
    const int E = in_sizes[2];         // 800000
    const int T = 4;

    // workspace layout (floats); h2 aliases eft (dead after scatter)
    float* ws   = (float*)d_ws;
    float* z    = ws;                          // N*64
    float* ssrc = z + (size_t)N * 64;          // N
    float* sdst = ssrc + N;                    // N
    float* abuf = sdst + N;                    // E
    float* mbuf = abuf + E;                    // N
    float* den  = mbuf + N;                    // N
    float* zphi = den + N;                     // 256
    float* beta = zphi + 256;                  // 16 (padded)
    float* h    = beta + 16;                   // N*T*96
    float* eft  = h + (size_t)N * T * 96;      // E*32
    float* h2   = eft;                         // N*T*64 <= E*32

    const int thr = 256;

    // z = node @ w_n  [N,128]x[128,64]
    gemm_f32_wmma<4><<<((N + 15) / 16 + 7) / 8, thr, 0, stream>>>(node, w_n, z, N, 128, 64);
    // eft = edges @ w_e  [E,64]x[64,32]
    gemm_f32_wmma<2><<<((E + 15) / 16 + 7) / 8, thr, 0, stream>>>(edgef, w_e, eft, E, 64, 32);
    // per-node attention scores
    node_score_kernel<<<(N + thr - 1) / thr, thr, 0, stream>>>(z, w_attn, ssrc, sdst, N);
    // zero h/den/zphi, m = -inf
    init_kernel<<<1024, thr, 0, stream>>>(h, mbuf, den, zphi, N * T * 96, N);
    // edge scores + segment max
    edge_score_kernel<<<(E + thr - 1) / thr, thr, 0, stream>>>(eft, src, dst, ssrc, sdst,
                                                               w_attn, abuf, mbuf, E);
    // exp + segment sum
    edge_exp_kernel<<<(E + thr - 1) / thr, thr, 0, stream>>>(dst, mbuf, abuf, den, E);
    // alpha-weighted scatter into h  (one wave per edge)
    edge_scatter_kernel<<<(E + 7) / 8, thr, 0, stream>>>(z, eft, abuf, den, src, dst, etype, h, E);
    // h2 = h @ w_fc2  [N*T,96]x[96,64]   (h2 aliases eft)
    gemm_f32_wmma<4><<<((N * T + 15) / 16 + 7) / 8, thr, 0, stream>>>(h, w_fc2, h2, N * T, 96, 64);
    // zphi = sum_n h2
    zphi_kernel<<<256, thr, 0, stream>>>(h2, zphi, N);
    // semantic attention weights
    beta_kernel<<<1, 32, 0, stream>>>(zphi, w_sem, beta);
    // Z = sum_t beta_t * h2[:,t,:]
    out_kernel<<<(N * 64 + thr - 1) / thr, thr, 0, stream>>>(h2, beta, out, N);
}